// QSScan_24524263260391
// MI455X (gfx1250) — compile-verified
//
#include <hip/hip_runtime.h>
#include <math.h>

// ---------------- problem constants ----------------
#define BATCH   2
#define DINNER  2048
#define SEQLEN  2048
#define DSTATE  16
#define NCH     (BATCH * DINNER)      // 4096 independent channels
#define NSEG    16                    // sequence segments (parallel scan decomposition)
#define SEGLEN  (SEQLEN / NSEG)       // 128 timesteps per segment
#define TL      32                    // timesteps per LDS subtile
#define NT      (SEGLEN / TL)         // 4 subtiles per segment
#define CPB     128                   // channels per block (4 waves @ wave32)
#define RS      36                    // LDS row stride (floats) for u/dt/z tiles (16B aligned, bank-spread)
#define BS      20                    // LDS row stride (floats) for transposed B/C tiles (16B aligned)

#define LOG2E 1.44269504088896340736f
#define LN2   0.69314718055994530942f

typedef int v4i __attribute__((vector_size(16)));

// ---------------- CDNA5 async global->LDS path ----------------
#ifndef __has_builtin
#define __has_builtin(x) 0
#endif

#if __has_builtin(__builtin_amdgcn_global_load_async_to_lds_b128) && \
    __has_builtin(__builtin_amdgcn_s_wait_asynccnt)
#define ASYNC_BUILTIN 1
#else
#define ASYNC_BUILTIN 0
#endif

// 16-byte async copy global -> LDS (per-lane addresses; tracked by ASYNCcnt)
__device__ __forceinline__ void async_ld16(void* lds, const void* g) {
#if ASYNC_BUILTIN
  __builtin_amdgcn_global_load_async_to_lds_b128(
      (v4i*)g, (v4i*)lds, 0, 0);
#else
  // ISA form: global_load_async_to_lds_b128 vdst(lds byte addr), v[addr:addr+1], off
  unsigned loff = (unsigned)(unsigned long long)lds;  // low 32 bits of flat LDS addr = LDS offset
  asm volatile("global_load_async_to_lds_b128 %0, %1, off"
               :: "v"(loff), "v"(g) : "memory");
#endif
}

template <int N>
__device__ __forceinline__ void wait_async() {
#if ASYNC_BUILTIN
  __builtin_amdgcn_s_wait_asynccnt(N);
#else
  asm volatile("s_wait_asynccnt %0" :: "i"(N) : "memory");
#endif
}

// ---------------- math helpers (map to v_exp_f32 / v_log_f32) ----------------
__device__ __forceinline__ float exp2fast(float x) { return __builtin_amdgcn_exp2f(x); }
__device__ __forceinline__ float log2fast(float x) { return __builtin_amdgcn_logf(x); }

__device__ __forceinline__ float softplus_f(float x) {
  float sp = LN2 * log2fast(1.0f + exp2fast(x * LOG2E));
  return x > 20.0f ? x : sp;
}
__device__ __forceinline__ float silu_f(float x) {
  return x / (1.0f + exp2fast(-x * LOG2E));
}

// ---------------- tile staging ----------------
// Stage a [CPB x TL] float tile of one (b,d,l)-major array into LDS.
// 8 B128 chunks per thread; consecutive tids hit consecutive 16B chunks -> coalesced.
__device__ __forceinline__ void issue_tile(const float* __restrict__ g,
                                           float* lds, int c0, int l, int tid) {
#pragma unroll
  for (int k = 0; k < 8; ++k) {
    int chunk = k * CPB + tid;
    int row   = chunk >> 3;
    int col   = (chunk & 7) << 2;  // float index
    size_t go = (size_t)(c0 + row) * SEQLEN + (size_t)l + col;
    async_ld16(&lds[row * RS + col], g + go);
  }
}

// Stage B or C for a whole segment, transposed to [t][n] (row stride BS), sync loads.
__device__ __forceinline__ void load_bc_seg(const float* __restrict__ g,
                                            float* lds, int bidx, int l0, int tid) {
#pragma unroll
  for (int n = 0; n < DSTATE; ++n) {
    float v = g[((size_t)bidx * DSTATE + n) * SEQLEN + l0 + tid];  // 128 consecutive floats
    lds[tid * BS + n] = v;
  }
}

// ---------------- kernel 1: per-segment partials (P = prod a, Q = local scan) ----------------
__global__ __launch_bounds__(CPB) void k1_partials(
    const float* __restrict__ u, const float* __restrict__ dt,
    const float* __restrict__ Bg, const float* __restrict__ A_log,
    const float* __restrict__ dt_bias,
    float* __restrict__ P, float* __restrict__ Q) {
  __shared__ __align__(16) float u_ld[2][CPB * RS];
  __shared__ __align__(16) float d_ld[2][CPB * RS];
  __shared__ __align__(16) float B_ld[SEGLEN * BS];

  const int tid  = threadIdx.x;
  const int c0   = blockIdx.x * CPB;
  const int c    = c0 + tid;
  const int bidx = c >> 11;            // c / DINNER
  const int d    = c & (DINNER - 1);
  const int s    = blockIdx.y;
  const int l0   = s * SEGLEN;

  float A2[DSTATE];
#pragma unroll
  for (int n = 0; n < DSTATE; ++n)
    A2[n] = -__expf(A_log[d * DSTATE + n]) * LOG2E;  // A*log2(e): a_t = exp2(delta*A2)
  const float dtb = dt_bias[d];

  load_bc_seg(Bg, B_ld, bidx, l0, tid);
  issue_tile(u,  u_ld[0], c0, l0, tid);
  issue_tile(dt, d_ld[0], c0, l0, tid);

  float x[DSTATE];
#pragma unroll
  for (int n = 0; n < DSTATE; ++n) x[n] = 0.0f;
  float dsum = 0.0f;

  for (int i = 0; i < NT; ++i) {
    const int cur = i & 1;
    if (i + 1 < NT) {
      issue_tile(u,  u_ld[cur ^ 1], c0, l0 + (i + 1) * TL, tid);
      issue_tile(dt, d_ld[cur ^ 1], c0, l0 + (i + 1) * TL, tid);
      wait_async<16>();   // 16 newer ops outstanding => current tile complete
    } else {
      wait_async<0>();
    }
    __syncthreads();

    const float* ur = &u_ld[cur][tid * RS];
    const float* dr = &d_ld[cur][tid * RS];
    for (int t4 = 0; t4 < TL; t4 += 4) {
      float4 u4 = *(const float4*)(ur + t4);
      float4 d4 = *(const float4*)(dr + t4);
      float uv_[4] = {u4.x, u4.y, u4.z, u4.w};
      float dv_[4] = {d4.x, d4.y, d4.z, d4.w};
#pragma unroll
      for (int j = 0; j < 4; ++j) {
        float delta = softplus_f(dv_[j] + dtb);
        dsum += delta;
        float du = delta * uv_[j];
        const float* Br = &B_ld[(i * TL + t4 + j) * BS];
        float4 B0 = *(const float4*)(Br + 0);
        float4 B1 = *(const float4*)(Br + 4);
        float4 B2 = *(const float4*)(Br + 8);
        float4 B3 = *(const float4*)(Br + 12);
        float Bv[DSTATE] = {B0.x, B0.y, B0.z, B0.w, B1.x, B1.y, B1.z, B1.w,
                            B2.x, B2.y, B2.z, B2.w, B3.x, B3.y, B3.z, B3.w};
#pragma unroll
        for (int n = 0; n < DSTATE; ++n) {
          float a = exp2fast(delta * A2[n]);
          x[n] = fmaf(a, x[n], du * Bv[n]);
        }
      }
    }
    __syncthreads();
  }

  const size_t base = ((size_t)s * NCH + c) * DSTATE;
#pragma unroll
  for (int n = 0; n < DSTATE; n += 4) {
    float4 q4 = {x[n], x[n + 1], x[n + 2], x[n + 3]};
    *(float4*)(Q + base + n) = q4;
    float4 p4 = {exp2fast(A2[n] * dsum),     exp2fast(A2[n + 1] * dsum),
                 exp2fast(A2[n + 2] * dsum), exp2fast(A2[n + 3] * dsum)};
    *(float4*)(P + base + n) = p4;
  }
}

// ---------------- kernel 2: scan over segments; write x_init in place over P ----------------
__global__ __launch_bounds__(256) void k2_seqscan(float* __restrict__ P,
                                                  const float* __restrict__ Q) {
  const int c = blockIdx.x * 256 + threadIdx.x;
  float x[DSTATE];
#pragma unroll
  for (int n = 0; n < DSTATE; ++n) x[n] = 0.0f;

  for (int s = 0; s < NSEG; ++s) {
    const size_t base = ((size_t)s * NCH + c) * DSTATE;
    float p[DSTATE], q[DSTATE];
#pragma unroll
    for (int n = 0; n < DSTATE; n += 4) {
      float4 p4 = *(const float4*)(P + base + n);
      float4 q4 = *(const float4*)(Q + base + n);
      p[n] = p4.x; p[n + 1] = p4.y; p[n + 2] = p4.z; p[n + 3] = p4.w;
      q[n] = q4.x; q[n + 1] = q4.y; q[n + 2] = q4.z; q[n + 3] = q4.w;
    }
#pragma unroll
    for (int n = 0; n < DSTATE; n += 4) {   // x_init for segment s overwrites P slot
      float4 xi = {x[n], x[n + 1], x[n + 2], x[n + 3]};
      *(float4*)(P + base + n) = xi;
    }
#pragma unroll
    for (int n = 0; n < DSTATE; ++n) x[n] = fmaf(p[n], x[n], q[n]);
  }
}

// ---------------- kernel 3: rerun recurrence from true init, emit output ----------------
__global__ __launch_bounds__(CPB) void k3_final(
    const float* __restrict__ u, const float* __restrict__ dt,
    const float* __restrict__ Bg, const float* __restrict__ Cg,
    const float* __restrict__ z, const float* __restrict__ A_log,
    const float* __restrict__ Dp, const float* __restrict__ dt_bias,
    const float* __restrict__ XI, float* __restrict__ out) {
  __shared__ __align__(16) float u_ld[2][CPB * RS];
  __shared__ __align__(16) float d_ld[2][CPB * RS];
  __shared__ __align__(16) float z_ld[2][CPB * RS];
  __shared__ __align__(16) float B_ld[SEGLEN * BS];
  __shared__ __align__(16) float C_ld[SEGLEN * BS];

  const int tid  = threadIdx.x;
  const int c0   = blockIdx.x * CPB;
  const int c    = c0 + tid;
  const int bidx = c >> 11;
  const int d    = c & (DINNER - 1);
  const int s    = blockIdx.y;
  const int l0   = s * SEGLEN;

  float A2[DSTATE];
#pragma unroll
  for (int n = 0; n < DSTATE; ++n)
    A2[n] = -__expf(A_log[d * DSTATE + n]) * LOG2E;
  const float dtb = dt_bias[d];
  const float Dd  = Dp[d];

  load_bc_seg(Bg, B_ld, bidx, l0, tid);
  load_bc_seg(Cg, C_ld, bidx, l0, tid);
  issue_tile(u,  u_ld[0], c0, l0, tid);
  issue_tile(dt, d_ld[0], c0, l0, tid);
  issue_tile(z,  z_ld[0], c0, l0, tid);

  float x[DSTATE];
  {
    const size_t base = ((size_t)s * NCH + c) * DSTATE;
#pragma unroll
    for (int n = 0; n < DSTATE; n += 4) {
      float4 xi = *(const float4*)(XI + base + n);
      x[n] = xi.x; x[n + 1] = xi.y; x[n + 2] = xi.z; x[n + 3] = xi.w;
    }
  }

  for (int i = 0; i < NT; ++i) {
    const int cur = i & 1;
    if (i + 1 < NT) {
      issue_tile(u,  u_ld[cur ^ 1], c0, l0 + (i + 1) * TL, tid);
      issue_tile(dt, d_ld[cur ^ 1], c0, l0 + (i + 1) * TL, tid);
      issue_tile(z,  z_ld[cur ^ 1], c0, l0 + (i + 1) * TL, tid);
      wait_async<24>();   // 24 newer ops outstanding => current tile complete
    } else {
      wait_async<0>();
    }
    __syncthreads();

    const float* ur = &u_ld[cur][tid * RS];
    const float* dr = &d_ld[cur][tid * RS];
    const float* zr = &z_ld[cur][tid * RS];
    for (int t4 = 0; t4 < TL; t4 += 4) {
      float4 u4 = *(const float4*)(ur + t4);
      float4 d4 = *(const float4*)(dr + t4);
      float4 z4 = *(const float4*)(zr + t4);
      float uv_[4] = {u4.x, u4.y, u4.z, u4.w};
      float dv_[4] = {d4.x, d4.y, d4.z, d4.w};
      float zv_[4] = {z4.x, z4.y, z4.z, z4.w};
#pragma unroll
      for (int j = 0; j < 4; ++j) {
        const int tg = i * TL + t4 + j;
        float delta = softplus_f(dv_[j] + dtb);
        float du = delta * uv_[j];
        const float* Br = &B_ld[tg * BS];
        float4 B0 = *(const float4*)(Br + 0);
        float4 B1 = *(const float4*)(Br + 4);
        float4 B2 = *(const float4*)(Br + 8);
        float4 B3 = *(const float4*)(Br + 12);
        float Bv[DSTATE] = {B0.x, B0.y, B0.z, B0.w, B1.x, B1.y, B1.z, B1.w,
                            B2.x, B2.y, B2.z, B2.w, B3.x, B3.y, B3.z, B3.w};
        const float* Cr = &C_ld[tg * BS];
        float4 C0 = *(const float4*)(Cr + 0);
        float4 C1 = *(const float4*)(Cr + 4);
        float4 C2 = *(const float4*)(Cr + 8);
        float4 C3 = *(const float4*)(Cr + 12);
        float Cv[DSTATE] = {C0.x, C0.y, C0.z, C0.w, C1.x, C1.y, C1.z, C1.w,
                            C2.x, C2.y, C2.z, C2.w, C3.x, C3.y, C3.z, C3.w};
        float y = 0.0f;
#pragma unroll
        for (int n = 0; n < DSTATE; ++n) {
          float a = exp2fast(delta * A2[n]);
          x[n] = fmaf(a, x[n], du * Bv[n]);
          y = fmaf(x[n], Cv[n], y);
        }
        y = fmaf(Dd, uv_[j], y) * silu_f(zv_[j]);
        out[((size_t)bidx * SEQLEN + (l0 + tg)) * DINNER + d] = y;  // lanes -> contiguous d
      }
    }
    __syncthreads();
  }
}

// ---------------- launcher ----------------
extern "C" void kernel_launch(void* const* d_in, const int* in_sizes, int n_in,
                              void* d_out, int out_size, void* d_ws, size_t ws_size,
                              hipStream_t stream) {
  const float* u   = (const float*)d_in[0];
  const float* dt  = (const float*)d_in[1];
  const float* Bg  = (const float*)d_in[2];
  const float* Cg  = (const float*)d_in[3];
  const float* z   = (const float*)d_in[4];
  const float* Al  = (const float*)d_in[5];
  const float* Dp  = (const float*)d_in[6];
  const float* dtb = (const float*)d_in[7];
  float* out = (float*)d_out;

  float* P = (float*)d_ws;                                 // [NSEG][NCH][DSTATE]
  float* Q = P + (size_t)NSEG * NCH * DSTATE;              // [NSEG][NCH][DSTATE]  (8 MB total)

  dim3 grid(NCH / CPB, NSEG), blk(CPB);
  k1_partials<<<grid, blk, 0, stream>>>(u, dt, Bg, Al, dtb, P, Q);
  k2_seqscan<<<dim3(NCH / 256), dim3(256), 0, stream>>>(P, Q);
  k3_final<<<grid, blk, 0, stream>>>(u, dt, Bg, Cg, z, Al, Dp, dtb, P, out);
}